// WassersteinLoss_51041391346371
// MI455X (gfx1250) — compile-verified
//
#include <hip/hip_runtime.h>
#include <hip/hip_bf16.h>
#include <stdint.h>

// ---------------------------------------------------------------------------
// WassersteinLoss on MI455X (gfx1250)
//   c = cumsum(pred - true, axis=1); out = 2/(T(T+1)) * mean_b( sum_t |c_t|*(T-t) )
// Memory-bound: 128 MB read -> ~5.5 us at 23.3 TB/s (and fits in 192 MB L2 on
// replay). One wave32 per row; serial carry across 128-elem chunks; 8-deep
// async global->LDS pipeline (ASYNCcnt) to hide HBM latency behind the serial
// scan dependency.
// ---------------------------------------------------------------------------

#define WAVES_PER_BLOCK 8
#define DEPTH           8          // async pipeline depth (power of two)
#define CHUNK           128        // elements per wave-chunk = 32 lanes * 4 floats
#define T_LEN           16384      // fixed by reference

__global__ void __launch_bounds__(WAVES_PER_BLOCK * 32)
wloss_row_scan(const float* __restrict__ pred,
               const float* __restrict__ tru,
               double* __restrict__ rowsum,
               int B)
{
    // [wave][buf][array(pred/true)][elem]  : 8*8*2*128*4B = 64 KB per block
    __shared__ __align__(16) float sbuf[WAVES_PER_BLOCK][DEPTH][2][CHUNK];

    const int lane = threadIdx.x & 31;
    const int w    = threadIdx.x >> 5;
    const int row  = blockIdx.x * WAVES_PER_BLOCK + w;
    if (row >= B) return;   // no barriers in this kernel -> early exit is safe

    const float* prow = pred + (size_t)row * T_LEN;
    const float* trow = tru  + (size_t)row * T_LEN;

    // Issue one 16B-per-lane async copy per array into this wave's LDS buffer.
    // VDST = per-lane LDS byte address (low 32 bits of generic shared ptr),
    // ADDR = per-lane 64-bit global address (GV mode).
    auto prefetch = [&](int chunk, int buf) {
        const unsigned lp = (unsigned)(unsigned long long)&sbuf[w][buf][0][lane * 4];
        const unsigned lt = (unsigned)(unsigned long long)&sbuf[w][buf][1][lane * 4];
        const unsigned long long gp =
            (unsigned long long)(prow + (size_t)chunk * CHUNK + lane * 4);
        const unsigned long long gt =
            (unsigned long long)(trow + (size_t)chunk * CHUNK + lane * 4);
        asm volatile("global_load_async_to_lds_b128 %0, %1, off"
                     :: "v"(lp), "v"(gp) : "memory");
        asm volatile("global_load_async_to_lds_b128 %0, %1, off"
                     :: "v"(lt), "v"(gt) : "memory");
    };

    float  carry = 0.0f;   // running cumsum entering the current chunk
    double acc   = 0.0;    // per-lane weighted |c| accumulator

    auto compute = [&](int buf, int cbase) {
        const float4 pv = *reinterpret_cast<const float4*>(&sbuf[w][buf][0][lane * 4]);
        const float4 tv = *reinterpret_cast<const float4*>(&sbuf[w][buf][1][lane * 4]);
        const float d0 = pv.x - tv.x;
        const float d1 = pv.y - tv.y;
        const float d2 = pv.z - tv.z;
        const float d3 = pv.w - tv.w;
        // lane-local inclusive prefix over 4 elements
        const float l0 = d0, l1 = l0 + d1, l2 = l1 + d2, l3 = l2 + d3;
        // wave32 inclusive scan of lane totals
        float incl = l3;
#pragma unroll
        for (int off = 1; off < 32; off <<= 1) {
            const float n = __shfl_up(incl, off, 32);
            if (lane >= off) incl += n;
        }
        const float cb = carry + (incl - l3);          // exclusive-scan base
        const int   t0 = cbase + lane * 4;             // global t index of elem 0
        const float w0 = (float)(T_LEN - t0);          // weight T - t
        float ca;
        ca  = fabsf(cb + l0) * w0;
        ca += fabsf(cb + l1) * (w0 - 1.0f);
        ca += fabsf(cb + l2) * (w0 - 2.0f);
        ca += fabsf(cb + l3) * (w0 - 3.0f);
        acc += (double)ca;
        carry += __shfl(incl, 31, 32);                 // chunk total -> carry
    };

    const int NC = T_LEN / CHUNK;   // 128 chunks per row

    // Prime the pipeline: DEPTH chunks in flight (2*DEPTH async loads).
#pragma unroll
    for (int k = 0; k < DEPTH; ++k) prefetch(k, k);

    // Steady state: chunk k is complete once ASYNCcnt <= 2*(DEPTH-1)
    // (async loads complete in order).
    for (int k = 0; k < NC - DEPTH; ++k) {
        asm volatile("s_wait_asynccnt %0" :: "n"(2 * (DEPTH - 1)) : "memory");
        const int buf = k & (DEPTH - 1);
        compute(buf, k * CHUNK);
        // Make sure our ds reads of this buffer fully completed before the
        // async DMA engine may overwrite it with chunk k+DEPTH.
        asm volatile("s_wait_dscnt 0" ::: "memory");
        prefetch(k + DEPTH, buf);
    }

    // Drain and finish the tail from LDS.
    asm volatile("s_wait_asynccnt 0" ::: "memory");
    for (int k = NC - DEPTH; k < NC; ++k)
        compute(k & (DEPTH - 1), k * CHUNK);

    // Wave32 reduction of per-lane accumulators; lane 0 writes the row sum.
#pragma unroll
    for (int off = 16; off; off >>= 1) acc += __shfl_down(acc, off, 32);
    if (lane == 0) rowsum[row] = acc;
}

__global__ void __launch_bounds__(256)
wloss_reduce(const double* __restrict__ rowsum, float* __restrict__ out,
             int B, double scale)
{
    __shared__ double s[256 / 32];
    double a = 0.0;
    for (int i = threadIdx.x; i < B; i += blockDim.x) a += rowsum[i];
#pragma unroll
    for (int off = 16; off; off >>= 1) a += __shfl_down(a, off, 32);
    const int w = threadIdx.x >> 5, lane = threadIdx.x & 31;
    if (lane == 0) s[w] = a;
    __syncthreads();
    if (threadIdx.x == 0) {
        double t = 0.0;
        for (int i = 0; i < (int)(blockDim.x / 32); ++i) t += s[i];
        out[0] = (float)(t * scale);   // fully overwrites d_out each call
    }
}

extern "C" void kernel_launch(void* const* d_in, const int* in_sizes, int n_in,
                              void* d_out, int out_size, void* d_ws, size_t ws_size,
                              hipStream_t stream)
{
    const float* pred = (const float*)d_in[0];
    const float* tru  = (const float*)d_in[1];
    const int T = T_LEN;
    const int B = in_sizes[0] / T;                  // 1024 for the reference

    double* rowsum = (double*)d_ws;                 // B * 8 bytes of scratch

    dim3 blk(WAVES_PER_BLOCK * 32);
    dim3 grd((B + WAVES_PER_BLOCK - 1) / WAVES_PER_BLOCK);
    hipLaunchKernelGGL(wloss_row_scan, grd, blk, 0, stream, pred, tru, rowsum, B);

    const double scale = 2.0 / ((double)T * (double)(T + 1) * (double)B);
    hipLaunchKernelGGL(wloss_reduce, dim3(1), dim3(256), 0, stream,
                       rowsum, (float*)d_out, B, scale);
}